// GraphGAT_73289321939356
// MI455X (gfx1250) — compile-verified
//
#include <hip/hip_runtime.h>
#include <hip/hip_bf16.h>

typedef __attribute__((ext_vector_type(2))) float v2f;
typedef __attribute__((ext_vector_type(8))) float v8f;

#define HEADS 4
#define CH    128
#define CHEAD 32

// ---------------------------------------------------------------------------
// atomic helpers
// ---------------------------------------------------------------------------
__device__ __forceinline__ void atomicMaxF32(float* addr, float value) {
  // sign-aware bit trick: correct across mixed signs, init with -3e38
  if (value >= 0.0f) {
    atomicMax((int*)addr, __float_as_int(value));
  } else {
    atomicMin((unsigned int*)addr, __float_as_uint(value));
  }
}

__device__ __forceinline__ void atomicAddF32(float* addr, float value) {
  unsafeAtomicAdd(addr, value);   // hardware global_atomic_add_f32
}

// ---------------------------------------------------------------------------
// FP32 WMMA GEMM:  H[N,128] = X[N,128] @ W[128,128]
// One wave owns a 16-row panel and ALL 8 column tiles simultaneously:
//   - 8 accumulator fragments (v8f c[8] = 64 VGPRs) stay register-resident
//     (touched by a WMMA every k-step -> allocator cannot profitably spill)
//   - k is the OUTER loop: the 2-VGPR A fragment is loaded once per k-step
//     and consumed by all 8 WMMAs -> full A reuse with no persistent array
//   - 8 independent accumulation chains per k-step pipeline the XDL latency
// A frag (16x4 f32): lane l holds rows m=l&15, k = k0 + 2*(l>>4) + {0,1}
// B frag (4x16 f32): lane l holds col n=l&15, k = k0 + 2*(l>>4) + {0,1}
// C/D (16x16 f32):  vgpr v, lane l -> row = v + 8*(l>>4), col = l&15
// ---------------------------------------------------------------------------
__global__ void __launch_bounds__(256, 1)
gat_gemm_wmma_f32(const float* __restrict__ X,
                  const float* __restrict__ W,
                  float* __restrict__ H, int N) {
  const int wave  = (int)((blockIdx.x * blockDim.x + threadIdx.x) >> 5);
  const int lane  = (int)(threadIdx.x & 31);
  const int row0  = wave * 16;
  if (row0 + 16 > N) return;               // wave-uniform: EXEC stays all-ones
  const int m     = lane & 15;
  const int khalf = lane >> 4;             // 0 or 1

  const float* xrow  = X + (size_t)(row0 + m) * CH;
  const float* wbase = W + m;

  v8f c[8];
  #pragma unroll
  for (int tn = 0; tn < 8; ++tn) c[tn] = (v8f){};

  #pragma unroll 4
  for (int i = 0; i < 32; ++i) {
    const int ka = i * 4 + khalf * 2;
    v2f a;
    a.x = xrow[ka];
    a.y = xrow[ka + 1];
    const float* wrow0 = wbase + (size_t)ka * CH;
    #pragma unroll
    for (int tn = 0; tn < 8; ++tn) {
      v2f b;
      b.x = wrow0[tn * 16];
      b.y = wrow0[CH + tn * 16];
      c[tn] = __builtin_amdgcn_wmma_f32_16x16x4_f32(
                  /*neg_a=*/false, a, /*neg_b=*/false, b,
                  /*c_mod=*/(short)0, c[tn], /*reuse_a=*/false, /*reuse_b=*/false);
    }
  }

  float* outbase = H + (size_t)(row0 + 8 * khalf) * CH + m;
  #pragma unroll
  for (int tn = 0; tn < 8; ++tn) {
    float* out = outbase + tn * 16;
    #pragma unroll
    for (int v = 0; v < 8; ++v) out[(size_t)v * CH] = c[tn][v];
  }
}

// ---------------------------------------------------------------------------
// attention logits: al_src[n,h] = sum_c h[n,h,c]*a_src[h,c]  (same for dst)
// one thread per (node, head); buf0 is L2-resident right after the GEMM
// ---------------------------------------------------------------------------
__global__ void gat_attn_logits(const float* __restrict__ H,
                                const float* __restrict__ a_src,
                                const float* __restrict__ a_dst,
                                float* __restrict__ al_s,
                                float* __restrict__ al_d, int N) {
  int idx = blockIdx.x * blockDim.x + threadIdx.x;
  if (idx >= N * HEADS) return;
  int n = idx >> 2, hd = idx & 3;
  const float* hp = H + (size_t)n * CH + hd * CHEAD;
  const float* as = a_src + hd * CHEAD;
  const float* ad = a_dst + hd * CHEAD;
  float s = 0.f, d = 0.f;
  #pragma unroll
  for (int c = 0; c < CHEAD; ++c) {
    float v = hp[c];
    s += v * as[c];
    d += v * ad[c];
  }
  al_s[idx] = s;
  al_d[idx] = d;
}

// ---------------------------------------------------------------------------
// per-layer init: zero aggregation output, -inf running max, zero denominator
// ---------------------------------------------------------------------------
__global__ void gat_init_layer(float* __restrict__ Hout,
                               float* __restrict__ mmax,
                               float* __restrict__ den, int N) {
  size_t i = (size_t)blockIdx.x * blockDim.x + threadIdx.x;
  if (i < (size_t)N * CH) Hout[i] = 0.0f;
  if (i < (size_t)N * HEADS) {
    mmax[i] = -3.0e38f;
    den[i]  = 0.0f;
  }
}

// ---------------------------------------------------------------------------
// edge pass A: e = leaky_relu(al_src[src] + al_dst[dst]); segment max
// one thread per edge (self-loops appended implicitly for k >= E)
// ---------------------------------------------------------------------------
__global__ void gat_edge_e_max(const int* __restrict__ srcI,
                               const int* __restrict__ dstI, int E, int N,
                               const float* __restrict__ al_s,
                               const float* __restrict__ al_d,
                               float* __restrict__ e_buf,
                               float* __restrict__ mmax) {
  int k = blockIdx.x * blockDim.x + threadIdx.x;
  if (k >= E + N) return;
  int s = (k < E) ? srcI[k] : (k - E);
  int d = (k < E) ? dstI[k] : (k - E);
  #pragma unroll
  for (int hd = 0; hd < HEADS; ++hd) {
    float e = al_s[s * HEADS + hd] + al_d[d * HEADS + hd];
    e = (e > 0.f) ? e : 0.2f * e;          // leaky_relu, slope 0.2
    e_buf[(size_t)k * HEADS + hd] = e;
    atomicMaxF32(&mmax[d * HEADS + hd], e);
  }
}

// ---------------------------------------------------------------------------
// edge pass B: ex = exp(e - m[dst]); segment sum of ex
// ---------------------------------------------------------------------------
__global__ void gat_edge_exp_den(const int* __restrict__ dstI, int E, int N,
                                 float* __restrict__ e_buf,       // in: e, out: ex
                                 const float* __restrict__ mmax,
                                 float* __restrict__ den) {
  int k = blockIdx.x * blockDim.x + threadIdx.x;
  if (k >= E + N) return;
  int d = (k < E) ? dstI[k] : (k - E);
  #pragma unroll
  for (int hd = 0; hd < HEADS; ++hd) {
    float ex = __expf(e_buf[(size_t)k * HEADS + hd] - mmax[d * HEADS + hd]);
    e_buf[(size_t)k * HEADS + hd] = ex;
    atomicAddF32(&den[d * HEADS + hd], ex);
  }
}

// ---------------------------------------------------------------------------
// edge pass C: Hout[dst] += (ex/den[dst]) * H[src]
// one WAVE per edge; each lane owns 4 channels (float4); head = lane/8
// ---------------------------------------------------------------------------
__global__ void gat_edge_aggregate(const int* __restrict__ srcI,
                                   const int* __restrict__ dstI, int E, int N,
                                   const float* __restrict__ ex_buf,
                                   const float* __restrict__ den,
                                   const float* __restrict__ H,
                                   float* __restrict__ Hout) {
  int wave = (int)((blockIdx.x * blockDim.x + threadIdx.x) >> 5);
  int lane = (int)(threadIdx.x & 31);
  if (wave >= E + N) return;
  int s = (wave < E) ? srcI[wave] : (wave - E);
  int d = (wave < E) ? dstI[wave] : (wave - E);
  int hd = lane >> 3;                                   // channels 4*lane..4*lane+3
  float alpha = ex_buf[(size_t)wave * HEADS + hd] / den[d * HEADS + hd];
  const float4 v = ((const float4*)(H + (size_t)s * CH))[lane];
  float* out = Hout + (size_t)d * CH + lane * 4;
  atomicAddF32(out + 0, v.x * alpha);
  atomicAddF32(out + 1, v.y * alpha);
  atomicAddF32(out + 2, v.z * alpha);
  atomicAddF32(out + 3, v.w * alpha);
}

// ---------------------------------------------------------------------------
// bias + ELU (in place)
// ---------------------------------------------------------------------------
__global__ void gat_bias_elu(float* __restrict__ H, const float* __restrict__ b,
                             int N) {
  size_t i = (size_t)blockIdx.x * blockDim.x + threadIdx.x;
  if (i >= (size_t)N * CH) return;
  int c = (int)(i & (CH - 1));
  float v = H[i] + b[c];
  H[i] = (v > 0.f) ? v : (__expf(v) - 1.0f);
}

// ---------------------------------------------------------------------------
// pooling: emb[c] = sum_n H[n,c]   (zero first, divide in head kernel)
// ---------------------------------------------------------------------------
__global__ void gat_zero_emb(float* __restrict__ emb) {
  emb[threadIdx.x] = 0.0f;
}

__global__ void gat_pool(const float* __restrict__ H, float* __restrict__ emb,
                         int N) {
  int c = threadIdx.x;                 // blockDim.x == CH, coalesced reads
  float acc = 0.0f;
  for (int n = blockIdx.x; n < N; n += gridDim.x)
    acc += H[(size_t)n * CH + c];
  atomicAddF32(&emb[c], acc);
}

// ---------------------------------------------------------------------------
// head: logits[o] = (emb_sum/N) @ head_w[:,o] + head_b[o]
// ---------------------------------------------------------------------------
__global__ void gat_head(const float* __restrict__ emb_sum,
                         const float* __restrict__ head_w,
                         const float* __restrict__ head_b,
                         float* __restrict__ out, int N) {
  int o = threadIdx.x;
  if (o >= 8) return;
  float invN = 1.0f / (float)N;
  float acc = head_b[o];
  #pragma unroll 8
  for (int c = 0; c < CH; ++c)
    acc += emb_sum[c] * invN * head_w[c * 8 + o];
  out[o] = acc;
}

// ---------------------------------------------------------------------------
extern "C" void kernel_launch(void* const* d_in, const int* in_sizes, int n_in,
                              void* d_out, int out_size, void* d_ws, size_t ws_size,
                              hipStream_t stream) {
  const float* x   = (const float*)d_in[0];
  const int*   ei  = (const int*)d_in[1];
  const int N = in_sizes[0] / CH;          // 100000
  const int E = in_sizes[1] / 2;           // 1600000
  const int* src = ei;
  const int* dst = ei + E;

  const float* Wl[3]  = {(const float*)d_in[2], (const float*)d_in[6],  (const float*)d_in[10]};
  const float* Asl[3] = {(const float*)d_in[3], (const float*)d_in[7],  (const float*)d_in[11]};
  const float* Adl[3] = {(const float*)d_in[4], (const float*)d_in[8],  (const float*)d_in[12]};
  const float* Bl[3]  = {(const float*)d_in[5], (const float*)d_in[9],  (const float*)d_in[13]};
  const float* head_w = (const float*)d_in[14];
  const float* head_b = (const float*)d_in[15];

  // workspace carve-up (all fp32)
  float* buf0  = (float*)d_ws;                          // [N,128] gemm output h
  float* buf1  = buf0 + (size_t)N * CH;                 // [N,128] aggregation out
  float* al_s  = buf1 + (size_t)N * CH;                 // [N,4]
  float* al_d  = al_s + (size_t)N * HEADS;              // [N,4]
  float* mmax  = al_d + (size_t)N * HEADS;              // [N,4]
  float* den   = mmax + (size_t)N * HEADS;              // [N,4]
  float* e_buf = den  + (size_t)N * HEADS;              // [E+N,4]
  float* emb   = e_buf + (size_t)(E + N) * HEADS;       // [128]

  const int Etot = E + N;
  const int thr  = 256;
  const int gemm_waves  = (N + 15) / 16;                // one wave per 16-row panel
  const int gemm_blocks = (gemm_waves + 7) / 8;         // 8 waves per 256-thr block
  const int nh_blocks   = (int)(((size_t)N * CH + thr - 1) / thr);
  const int nheads_blk  = (N * HEADS + thr - 1) / thr;
  const int edge_blk    = (Etot + thr - 1) / thr;
  const int agg_blocks  = (int)(((size_t)Etot * 32 + thr - 1) / thr);

  const float* cur = x;
  for (int l = 0; l < 3; ++l) {
    gat_gemm_wmma_f32<<<gemm_blocks, thr, 0, stream>>>(cur, Wl[l], buf0, N);
    gat_attn_logits<<<nheads_blk, thr, 0, stream>>>(buf0, Asl[l], Adl[l], al_s, al_d, N);
    gat_init_layer<<<nh_blocks, thr, 0, stream>>>(buf1, mmax, den, N);
    gat_edge_e_max<<<edge_blk, thr, 0, stream>>>(src, dst, E, N, al_s, al_d, e_buf, mmax);
    gat_edge_exp_den<<<edge_blk, thr, 0, stream>>>(dst, E, N, e_buf, mmax, den);
    gat_edge_aggregate<<<agg_blocks, thr, 0, stream>>>(src, dst, E, N, e_buf, den, buf0, buf1);
    gat_bias_elu<<<nh_blocks, thr, 0, stream>>>(buf1, Bl[l], N);
    cur = buf1;
  }

  gat_zero_emb<<<1, CH, 0, stream>>>(emb);
  gat_pool<<<512, CH, 0, stream>>>(cur, emb, N);
  gat_head<<<1, 64, 0, stream>>>(emb, head_w, head_b, (float*)d_out, N);
}